// finetune_ENDE_74062416052579
// MI455X (gfx1250) — compile-verified
//
#include <hip/hip_runtime.h>
#include <hip/hip_bf16.h>
#include <math.h>

// ---------------------------------------------------------------------------
// Fused contrastive (CTL) loss on MI455X / gfx1250.
//   B = 4096 rows, D = 576 features, fp32 in, fp32 scalar loss out.
//
// Design (see analysis): row-streaming GEMM via V_WMMA_F32_16X16X4_F32 with
// the ctl elementwise transform + online logsumexp fused per 16x16 tile; the
// [B,2B] logits matrix is never materialized. Inputs (19 MB) are L2-resident.
// ---------------------------------------------------------------------------

typedef float v2f __attribute__((ext_vector_type(2)));
typedef float v8f __attribute__((ext_vector_type(8)));

#define CTL_B     4096
#define CTL_D     576
#define CTL_DYN   0.9f
#define CHUNK     512          // columns (of the 2B concat space) per workgroup
#define NCHUNK    16           // 8192 / CHUNK
#define LDS_STRIDE 580         // 576 + 4 pad -> rows hit distinct LDS bank groups

// ---------------------------------------------------------------------------
// Kernel 1: per-row stats.  qqs[b]=|src_b|^2, qqt[b]=|trg_b|^2,
//           dpos[b]=||src_b - trg_b||.   One wave32 per row.
// ---------------------------------------------------------------------------
__global__ __launch_bounds__(32)
void ctl_row_stats(const float* __restrict__ src, const float* __restrict__ trg,
                   float* __restrict__ qqs, float* __restrict__ qqt,
                   float* __restrict__ dpos)
{
    const int b    = blockIdx.x;
    const int lane = threadIdx.x;
    const float* s = src + (size_t)b * CTL_D;
    const float* t = trg + (size_t)b * CTL_D;

    float ss = 0.f, tt = 0.f, dd = 0.f;
    for (int k = lane; k < CTL_D; k += 32) {
        float sv = s[k], tv = t[k];
        ss += sv * sv;
        tt += tv * tv;
        float dv = sv - tv;
        dd += dv * dv;
    }
#pragma unroll
    for (int off = 16; off > 0; off >>= 1) {
        ss += __shfl_xor(ss, off, 32);
        tt += __shfl_xor(tt, off, 32);
        dd += __shfl_xor(dd, off, 32);
    }
    if (lane == 0) {
        qqs[b]  = ss;
        qqt[b]  = tt;
        dpos[b] = sqrtf(dd);
    }
}

// ---------------------------------------------------------------------------
// Kernel 2: main fused GEMM + ctl transform + online logsumexp partials.
//   grid = (256 row-blocks of 16, 16 column-chunks of 512), block = 1 wave32.
//   Chunks 0..7  -> logits_st rows (q = src, keys = trg)
//   Chunks 8..15 -> logits_ts rows (q = trg, keys = src, -1e9 on diagonal)
//   Writes (rowmax, sumexp) per (row, chunk) to workspace.
// ---------------------------------------------------------------------------
__global__ __launch_bounds__(32)
void ctl_partial(const float* __restrict__ src, const float* __restrict__ trg,
                 const float* __restrict__ qqs, const float* __restrict__ qqt,
                 const float* __restrict__ dpos,
                 float* __restrict__ pmax, float* __restrict__ psum)
{
    __shared__ float As[16 * LDS_STRIDE];     // 37,120 B of the 320 KB WGP LDS

    const int rb   = blockIdx.x;              // row block: rows [16rb, 16rb+16)
    const int ch   = blockIdx.y;              // chunk 0..15
    const int lane = threadIdx.x;
    const int lo   = lane & 15;               // row (A) / col (B) within tile
    const int hi   = lane >> 4;               // K sub-offset selector

    const bool  isTs  = ch >= 8;
    const float* Q    = isTs ? trg : src;
    const float* Km   = isTs ? src : trg;
    const float* qqA  = isTs ? qqt : qqs;     // |q_b|^2 for this half
    const float* kkA  = isTs ? qqs : qqt;     // |k_j|^2 for this half
    const int    m0   = rb * 16;
    const int    j0b  = (ch & 7) * CHUNK;     // key-index base within half

    // Stage the 16 q-rows into LDS (padded stride -> conflict-free b64 reads).
    for (int m = 0; m < 16; ++m) {
        const float* qrow = Q + (size_t)(m0 + m) * CTL_D;
        for (int k = lane * 4; k < CTL_D; k += 128) {
            *(float4*)&As[m * LDS_STRIDE + k] = *(const float4*)&qrow[k];
        }
    }
    __syncthreads();

    // Per-lane row constants: this lane's 8 accumulator slots cover rows
    // m = v + 8*hi  (WMMA f32 16x16 C/D layout).
    float qq_r[8], dp_r[8];
#pragma unroll
    for (int v = 0; v < 8; ++v) {
        const int mg = m0 + v + 8 * hi;
        qq_r[v] = qqA[mg];
        dp_r[v] = dpos[mg];
    }

    float rmax[8], rsum[8];
#pragma unroll
    for (int v = 0; v < 8; ++v) { rmax[v] = -3.0e38f; rsum[v] = 0.f; }

    // 32 column tiles of 16 keys each.
    for (int t = 0; t < CHUNK / 16; ++t) {
        const int jg = j0b + t * 16 + lo;             // this lane's key index
        const float* brow = Km + (size_t)jg * CTL_D;  // B column n=lo
        const float* arow = &As[lo * LDS_STRIDE];     // A row    m=lo
        const int koff = hi * 2;

        v8f c = {0.f, 0.f, 0.f, 0.f, 0.f, 0.f, 0.f, 0.f};
#pragma unroll 4
        for (int k0 = 0; k0 < CTL_D; k0 += 4) {
            v2f a = *(const v2f*)&arow[k0 + koff];    // A[m][k0+koff .. +1]
            v2f b = *(const v2f*)&brow[k0 + koff];    // B[k0+koff..][n]
            c = __builtin_amdgcn_wmma_f32_16x16x4_f32(
                    false, a, false, b, (short)0, c, false, false);
        }

        // ctl transform + online (max,sum) update.  All 8 slots share column
        // n = lo, so one kk load per tile per lane.
        const float kkv = kkA[jg];
#pragma unroll
        for (int v = 0; v < 8; ++v) {
            const float qk  = c[v];
            const float qqv = qq_r[v];
            const float dp  = dp_r[v];
            float sq   = fmaxf(qqv + kkv - 2.f * qk, 1e-12f);
            float dneg = sqrtf(sq);
            // lam = (dpos/dneg)^0.9 ; dpos==0 -> log(0)=-inf -> lam=0 (matches ref)
            float lam  = __expf(CTL_DYN * __logf(dp / dneg));
            float logit = (dneg > dp) ? (qqv + lam * (qk - qqv)) : qk;
            if (isTs && jg == (m0 + v + 8 * hi)) logit += -1.0e9f;

            float mnew = fmaxf(rmax[v], logit);
            rsum[v] = rsum[v] * __expf(rmax[v] - mnew) + __expf(logit - mnew);
            rmax[v] = mnew;
        }
    }

    // Merge (max,sum) across the 16 lanes that share each row
    // (xor offsets 1,2,4,8 stay within each 16-lane group).
#pragma unroll
    for (int off = 1; off < 16; off <<= 1) {
#pragma unroll
        for (int v = 0; v < 8; ++v) {
            float om = __shfl_xor(rmax[v], off, 32);
            float os = __shfl_xor(rsum[v], off, 32);
            float mnew = fmaxf(rmax[v], om);
            rsum[v] = rsum[v] * __expf(rmax[v] - mnew) + os * __expf(om - mnew);
            rmax[v] = mnew;
        }
    }

    if (lo == 0) {
#pragma unroll
        for (int v = 0; v < 8; ++v) {
            const int mg = m0 + v + 8 * hi;
            pmax[mg * NCHUNK + ch] = rmax[v];
            psum[mg * NCHUNK + ch] = rsum[v];
        }
    }
}

// ---------------------------------------------------------------------------
// Kernel 3: combine chunk partials per row, subtract diagonal logit
//   logits_st[b,b] = <src_b,trg_b> = (qqs+qqt-dpos^2)/2, and mean-reduce.
// ---------------------------------------------------------------------------
__global__ __launch_bounds__(256)
void ctl_finalize(const float* __restrict__ qqs, const float* __restrict__ qqt,
                  const float* __restrict__ dpos,
                  const float* __restrict__ pmax, const float* __restrict__ psum,
                  float* __restrict__ out)
{
    __shared__ float red[256];
    const int tid = threadIdx.x;

    float acc = 0.f;
    for (int r = 0; r < CTL_B / 256; ++r) {
        const int b = tid + r * 256;
        float M = -3.0e38f;
#pragma unroll
        for (int c = 0; c < NCHUNK; ++c) M = fmaxf(M, pmax[b * NCHUNK + c]);
        float S = 0.f;
#pragma unroll
        for (int c = 0; c < NCHUNK; ++c)
            S += psum[b * NCHUNK + c] * __expf(pmax[b * NCHUNK + c] - M);
        float lse  = __logf(S) + M;
        float diag = 0.5f * (qqs[b] + qqt[b] - dpos[b] * dpos[b]);
        acc += lse - diag;
    }

    red[tid] = acc;
    __syncthreads();
#pragma unroll
    for (int s = 128; s > 0; s >>= 1) {
        if (tid < s) red[tid] += red[tid + s];
        __syncthreads();
    }
    if (tid == 0) out[0] = red[0] / (float)CTL_B;
}

// ---------------------------------------------------------------------------
extern "C" void kernel_launch(void* const* d_in, const int* in_sizes, int n_in,
                              void* d_out, int out_size, void* d_ws, size_t ws_size,
                              hipStream_t stream)
{
    (void)in_sizes; (void)n_in; (void)out_size; (void)ws_size;
    const float* src = (const float*)d_in[0];
    const float* trg = (const float*)d_in[1];
    float* out = (float*)d_out;

    // Workspace layout (floats): qqs[4096] | qqt[4096] | dpos[4096]
    //                            | pmax[4096*16] | psum[4096*16]   (~560 KB)
    float* ws   = (float*)d_ws;
    float* qqs  = ws;
    float* qqt  = ws + CTL_B;
    float* dpos = ws + 2 * CTL_B;
    float* pmax = ws + 3 * CTL_B;
    float* psum = pmax + CTL_B * NCHUNK;

    ctl_row_stats<<<CTL_B, 32, 0, stream>>>(src, trg, qqs, qqt, dpos);
    ctl_partial<<<dim3(CTL_B / 16, NCHUNK), 32, 0, stream>>>(
        src, trg, qqs, qqt, dpos, pmax, psum);
    ctl_finalize<<<1, 256, 0, stream>>>(qqs, qqt, dpos, pmax, psum, out);
}